// Model_91036126806521
// MI455X (gfx1250) — compile-verified
//
#include <hip/hip_runtime.h>
#include <stdint.h>

// Chunked gated linear-attention scan for MI455X (gfx1250, wave32).
// - f32 WMMA (v_wmma_f32_16x16x4_f32): reference-matching fp32 math; kernel is
//   memory-bound (~200MB traffic -> ~9us floor @ 23.3 TB/s), so f32 MACs are free.
// - Double-buffered GLOBAL_LOAD_ASYNC_TO_LDS (ASYNCcnt) hides per-chunk HBM
//   latency behind the WMMA phases of the serial T=64 scan.
// - Non-temporal output stores keep L2 free for w/k re-reads across V-tiles.

typedef __attribute__((ext_vector_type(2))) float v2f;
typedef __attribute__((ext_vector_type(4))) float v4f;
typedef __attribute__((ext_vector_type(8))) float v8f;

#define Bb    2
#define Ss    4096
#define Hh    16
#define KD    64
#define VD    128
#define CHUNK 64
#define Tt    (Ss / CHUNK)            // 64 sequential chunk steps
#define VT    32                      // V tile per block
#define NBLK  (Bb * Hh * (VD / VT))   // 128 blocks

// padded LDS row strides (floats): conflict-friendly, 16B-aligned rows
#define WPAD 68                       // 64 + 4
#define VPAD 36                       // 32 + 4

// each thread issues exactly 11 async ops per chunk group (8 w/k + 2 u + 1 g)
#define ASYNC_GROUP 11

__device__ __forceinline__ uint32_t lds_off(const void* p) {
    // generic pointers to LDS carry the workgroup-LDS byte offset in addr[31:0]
    return (uint32_t)(uintptr_t)p;
}
__device__ __forceinline__ void async_b128(const void* g, void* l) {
    asm volatile("global_load_async_to_lds_b128 %0, %1, off"
                 :: "v"(lds_off(l)), "v"(g) : "memory");
}
__device__ __forceinline__ void async_b32(const void* g, void* l) {
    asm volatile("global_load_async_to_lds_b32 %0, %1, off"
                 :: "v"(lds_off(l)), "v"(g) : "memory");
}

__global__ __launch_bounds__(256) void gla_chunk_scan(
    const float* __restrict__ kin, const float* __restrict__ win,
    const float* __restrict__ uin, const float* __restrict__ gin,
    float* __restrict__ out)
{
    __shared__ __align__(16) float lds_w[2][CHUNK][WPAD];  // w chunk, [c][k]
    __shared__ __align__(16) float lds_k[2][CHUNK][WPAD];  // k chunk, [c][k]
    __shared__ __align__(16) float lds_u[2][CHUNK][VPAD];  // u tile,  [c][v]
    __shared__ __align__(16) float lds_g[2][CHUNK];
    __shared__ __align__(16) float lds_bv[CHUNK][VPAD];    // bv tile, [c][v]
    __shared__ __align__(16) float lds_h[KD][VPAD];        // state at chunk entry, [k][v]

    const int blk = blockIdx.x;
    const int vt  = blk % (VD / VT);
    const int bh  = blk / (VD / VT);
    const int h   = bh % Hh;
    const int b   = bh / Hh;
    const int v0  = vt * VT;

    const int tid  = threadIdx.x;
    const int lane = tid & 31;
    const int wave = tid >> 5;        // 0..7
    const int mi   = wave >> 1;       // M tile 0..3  (rows of 16)
    const int ni   = wave & 1;        // N tile 0..1  (cols of 16)
    const int half = lane >> 4;       // lane half (0/1)
    const int l16  = lane & 15;

    // per-thread source coordinates for the async copy group
    const int cc = tid >> 2;               // chunk row 0..63 for w/k/u
    const int q  = (tid & 3) << 4;         // w/k col start (16 floats)
    const int qv = (tid & 3) << 3;         // u col start (8 floats)
    const int cg = tid & 63;               // g element (replicated x4 lanes)

    // ---- issue async copy group for one chunk into buffer `buf` ----
    auto issue_group = [&](int buf, int t) {
        const size_t rowbase = (size_t)((b * Ss + t * CHUNK + cc) * Hh + h);
        const size_t wkbase  = rowbase * KD + q;
        #pragma unroll
        for (int j = 0; j < 4; ++j) {
            async_b128(win + wkbase + 4 * j, &lds_w[buf][cc][q + 4 * j]);
            async_b128(kin + wkbase + 4 * j, &lds_k[buf][cc][q + 4 * j]);
        }
        const size_t ub = rowbase * VD + v0 + qv;
        async_b128(uin + ub,     &lds_u[buf][cc][qv]);
        async_b128(uin + ub + 4, &lds_u[buf][cc][qv + 4]);
        async_b32(gin + (size_t)((b * Ss + t * CHUNK + cg) * Hh + h),
                  &lds_g[buf][cg]);
    };

    // zero the shared state (state at entry of chunk 0)
    for (int i = tid; i < KD * VPAD; i += 256) ((float*)lds_h)[i] = 0.0f;

    // persistent per-wave state tile (h) kept in registers across the scan
    v8f hacc = {0.f, 0.f, 0.f, 0.f, 0.f, 0.f, 0.f, 0.f};

    // prologue: start chunk 0 copies
    issue_group(0, 0);

    __syncthreads();

    for (int t = 0; t < Tt; ++t) {
        const int cur = t & 1;

        // start next chunk's copies, then wait for current chunk's group
        if (t + 1 < Tt) {
            issue_group(cur ^ 1, t + 1);
            asm volatile("s_wait_asynccnt %0" :: "n"(ASYNC_GROUP) : "memory");
        } else {
            asm volatile("s_wait_asynccnt 0" ::: "memory");
        }
        __syncthreads();   // all waves' current-buffer data resident in LDS

        const float gl = lds_g[cur][CHUNK - 1];

        // ---- phase 1: acc = (w @ h) tile via 16x f32 WMMA, then bv ----
        v8f acc = {0.f, 0.f, 0.f, 0.f, 0.f, 0.f, 0.f, 0.f};
        const int mrow = mi * 16 + l16;
        const int ncol = ni * 16 + l16;
        #pragma unroll
        for (int kk = 0; kk < KD; kk += 4) {
            const int kidx = kk + 2 * half;
            v2f a, bf;
            a[0]  = lds_w[cur][mrow][kidx];
            a[1]  = lds_w[cur][mrow][kidx + 1];
            bf[0] = lds_h[kidx][ncol];
            bf[1] = lds_h[kidx + 1][ncol];
            acc = __builtin_amdgcn_wmma_f32_16x16x4_f32(
                /*neg_a=*/false, a, /*neg_b=*/false, bf,
                /*c_mod=*/(short)0, acc, /*reuse_a=*/false, /*reuse_b=*/false);
        }
        #pragma unroll
        for (int r = 0; r < 8; ++r) {
            const int row = mi * 16 + r + 8 * half;       // chunk position c
            const float sc = __expf(gl - lds_g[cur][row]);
            lds_bv[row][ncol] = (lds_u[cur][row][ncol] - acc[r]) * sc;
        }

        // ---- emit h_out = state at chunk entry (streaming, non-temporal) ----
        {
            const int kk = tid >> 2;
            const size_t ob = (((size_t)((b * Tt + t) * Hh + h)) * KD + kk) * VD + v0 + qv;
            v4f o0 = *(const v4f*)&lds_h[kk][qv];
            v4f o1 = *(const v4f*)&lds_h[kk][qv + 4];
            __builtin_nontemporal_store(o0, (v4f*)(out + ob));
            __builtin_nontemporal_store(o1, (v4f*)(out + ob + 4));
        }
        __syncthreads();   // bv complete + all lds_h readers done before update

        // ---- phase 2: h = h*exp(gl) + k^T @ bv, tile in registers ----
        const float egl = __expf(gl);
        #pragma unroll
        for (int r = 0; r < 8; ++r) hacc[r] *= egl;
        #pragma unroll
        for (int c0 = 0; c0 < CHUNK; c0 += 4) {
            const int cidx = c0 + 2 * half;
            v2f a, bf;
            a[0]  = lds_k[cur][cidx][mrow];       // A = k^T : A[kdim][c]
            a[1]  = lds_k[cur][cidx + 1][mrow];
            bf[0] = lds_bv[cidx][ncol];
            bf[1] = lds_bv[cidx + 1][ncol];
            hacc = __builtin_amdgcn_wmma_f32_16x16x4_f32(
                false, a, false, bf, (short)0, hacc, false, false);
        }
        // publish new state for next chunk's phase 1 / h_out
        #pragma unroll
        for (int r = 0; r < 8; ++r)
            lds_h[mi * 16 + r + 8 * half][ncol] = hacc[r];
        __syncthreads();
    }
}

extern "C" void kernel_launch(void* const* d_in, const int* in_sizes, int n_in,
                              void* d_out, int out_size, void* d_ws, size_t ws_size,
                              hipStream_t stream) {
    const float* k = (const float*)d_in[0];
    const float* w = (const float*)d_in[1];
    const float* u = (const float*)d_in[2];
    const float* g = (const float*)d_in[3];
    float* out = (float*)d_out;
    (void)in_sizes; (void)n_in; (void)out_size; (void)d_ws; (void)ws_size;
    gla_chunk_scan<<<dim3(NBLK), dim3(256), 0, stream>>>(k, w, u, g, out);
}